// GAT_19000935317531
// MI455X (gfx1250) — compile-verified
//
#include <hip/hip_runtime.h>
#include <hip/hip_bf16.h>

#define NUM_USERS 50000
#define NUM_ITEMS 20000
#define NNODES    70000           // 4375 * 16
#define MTILES    (NNODES / 16)   // 4375
#define HID       128
#define NE        1000000
#define EPLUS     (NE + NNODES)   // edges + self loops = 1,070,000
#define NEG_SLOPE 0.2f

typedef float v2f __attribute__((ext_vector_type(2)));
typedef float v8f __attribute__((ext_vector_type(8)));

// ---- order-preserving float <-> uint key (for native atomicMax segment-max) ----
__device__ __forceinline__ unsigned fkey(float f) {
    unsigned b = __float_as_uint(f);
    return (b & 0x80000000u) ? ~b : (b | 0x80000000u);
}
__device__ __forceinline__ float funkey(unsigned k) {
    unsigned b = (k & 0x80000000u) ? (k & 0x7FFFFFFFu) : ~k;
    return __uint_as_float(b);
}

// ---- concat users/items embeddings into x0 ----
__global__ __launch_bounds__(256) void concat_k(const float* __restrict__ u,
                                                const float* __restrict__ it,
                                                float* __restrict__ x) {
    int i = blockIdx.x * 256 + threadIdx.x;
    const int UW = NUM_USERS * HID;
    if (i < NNODES * HID) x[i] = (i < UW) ? u[i] : it[i - UW];
}

// ---- h = x @ W  via V_WMMA_F32_16X16X4_F32 ----
// One wave owns a full 16x128 output row-block: 8 v8f accumulators, each A
// fetch (v2f) feeds 8 WMMAs. A (36 MB/layer) is read exactly once; W (64 KB)
// streams from L2. Block = 8 waves = 8 consecutive M tiles.
__global__ __launch_bounds__(256) void gemm_wmma(const float* __restrict__ X,
                                                 const float* __restrict__ Wl,
                                                 float* __restrict__ H) {
    const int wave  = threadIdx.x >> 5;
    const int lane  = threadIdx.x & 31;
    const int tileM = blockIdx.x * 8 + wave;     // wave-uniform guard: EXEC stays all-1s
    if (tileM >= MTILES) return;
    const int m     = lane & 15;
    const int khalf = (lane >> 4) * 2;           // 0 or 2 (K sub-offset per lane half)
    const int row   = tileM * 16 + m;

    v8f c[8] = {};
    const float* xr = X + (size_t)row * HID + khalf;     // A: lane's 2 K values
    const float* wr = Wl + (size_t)khalf * HID + m;      // B: lane = N within tile

    for (int k = 0; k < HID; k += 4) {
        // A 16x4: lanes0-15 K=k,k+1 ; lanes16-31 K=k+2,k+3
        v2f a = *(const v2f*)(xr + k);
        const float* w0 = wr + (size_t)k * HID;          // row k+khalf
        const float* w1 = w0 + HID;                      // row k+khalf+1
#pragma unroll
        for (int t = 0; t < 8; ++t) {
            v2f b;
            b.x = w0[t * 16];
            b.y = w1[t * 16];
            c[t] = __builtin_amdgcn_wmma_f32_16x16x4_f32(false, a, false, b,
                                                         (short)0, c[t], false, false);
        }
    }
    // D 16x16 f32: VGPR r -> M = r (lanes 0-15) / r+8 (lanes 16-31), N = lane&15
    const int rbase = tileM * 16 + ((lane >> 4) ? 8 : 0);
#pragma unroll
    for (int t = 0; t < 8; ++t)
#pragma unroll
        for (int r = 0; r < 8; ++r)
            H[(size_t)(rbase + r) * HID + t * 16 + m] = c[t][r];
}

// ---- attention logits: as[n] = h[n].att_src, ad[n] = h[n].att_dst (wave/node) ----
__global__ __launch_bounds__(256) void alpha_k(const float* __restrict__ H,
                                               const float* __restrict__ Asrc,
                                               const float* __restrict__ Adst,
                                               float* __restrict__ as_out,
                                               float* __restrict__ ad_out) {
    const int warp = threadIdx.x >> 5, lane = threadIdx.x & 31;
    const int node = blockIdx.x * 8 + warp;
    if (node >= NNODES) return;
    float4 hv = ((const float4*)H)[(size_t)node * 32 + lane];
    float4 av = ((const float4*)Asrc)[lane];
    float4 dv = ((const float4*)Adst)[lane];
    float ss = hv.x * av.x + hv.y * av.y + hv.z * av.z + hv.w * av.w;
    float sd = hv.x * dv.x + hv.y * dv.y + hv.z * dv.z + hv.w * dv.w;
#pragma unroll
    for (int off = 16; off > 0; off >>= 1) {
        ss += __shfl_xor(ss, off, 32);
        sd += __shfl_xor(sd, off, 32);
    }
    if (lane == 0) { as_out[node] = ss; ad_out[node] = sd; }
}

// ---- per-node state init: segment-max key = 0 (== -inf key floor), denom = 0 ----
__global__ __launch_bounds__(256) void init_nodes(unsigned* __restrict__ mkey,
                                                  float* __restrict__ den) {
    int i = blockIdx.x * 256 + threadIdx.x;
    if (i < NNODES) { mkey[i] = 0u; den[i] = 0.f; }
}

// ---- init aggregation target to broadcast bias (folds the +bias epilogue) ----
__global__ __launch_bounds__(256) void init_bias(float* __restrict__ out,
                                                 const float* __restrict__ b) {
    int i = blockIdx.x * 256 + threadIdx.x;
    if (i < NNODES * HID) out[i] = b[i & (HID - 1)];
}

__device__ __forceinline__ void edge_sd(int i, const int* __restrict__ ei,
                                        int& s, int& d) {
    if (i < NE) { s = ei[i]; d = ei[NE + i]; }
    else        { s = d = i - NE; }
}

// ---- pass 1: segment max of leaky_relu logits via native uint atomicMax ----
__global__ __launch_bounds__(256) void edge_max_k(const int* __restrict__ ei,
                                                  const float* __restrict__ as_,
                                                  const float* __restrict__ ad_,
                                                  unsigned* __restrict__ mkey) {
    int i = blockIdx.x * 256 + threadIdx.x;
    if (i >= EPLUS) return;
    int s, d; edge_sd(i, ei, s, d);
    float e = as_[s] + ad_[d];
    e = (e >= 0.f) ? e : NEG_SLOPE * e;
    atomicMax(mkey + d, fkey(e));
}

// ---- pass 2: ex = exp(e - m[d]); store ex; denom[d] += ex ----
__global__ __launch_bounds__(256) void edge_exp_k(const int* __restrict__ ei,
                                                  const float* __restrict__ as_,
                                                  const float* __restrict__ ad_,
                                                  const unsigned* __restrict__ mkey,
                                                  float* __restrict__ den,
                                                  float* __restrict__ exb) {
    int i = blockIdx.x * 256 + threadIdx.x;
    if (i >= EPLUS) return;
    int s, d; edge_sd(i, ei, s, d);
    float e = as_[s] + ad_[d];
    e = (e >= 0.f) ? e : NEG_SLOPE * e;
    float ex = expf(e - funkey(mkey[d]));
    exb[i] = ex;
    atomicAdd(den + d, ex);
}

// ---- pass 3: out[d] += alpha * h[s]; one wave per edge, float4 per lane ----
__global__ __launch_bounds__(256) void edge_agg_k(const int* __restrict__ ei,
                                                  const float* __restrict__ H,
                                                  const float* __restrict__ exb,
                                                  const float* __restrict__ den,
                                                  float* __restrict__ out) {
    const int warp = threadIdx.x >> 5, lane = threadIdx.x & 31;
    const int e = blockIdx.x * 8 + warp;
    if (e >= EPLUS) return;
    int s, d; edge_sd(e, ei, s, d);
    float alpha = exb[e] / den[d];
    float4 hv = ((const float4*)H)[(size_t)s * 32 + lane];
    float* o = out + (size_t)d * HID + lane * 4;
    atomicAdd(o + 0, hv.x * alpha);
    atomicAdd(o + 1, hv.y * alpha);
    atomicAdd(o + 2, hv.z * alpha);
    atomicAdd(o + 3, hv.w * alpha);
}

extern "C" void kernel_launch(void* const* d_in, const int* in_sizes, int n_in,
                              void* d_out, int out_size, void* d_ws, size_t ws_size,
                              hipStream_t stream) {
    const int*   ei      = (const int*)d_in[0];
    const float* users   = (const float*)d_in[1];
    const float* items   = (const float*)d_in[2];
    const float* W       = (const float*)d_in[3];
    const float* att_src = (const float*)d_in[4];
    const float* att_dst = (const float*)d_in[5];
    const float* bias    = (const float*)d_in[6];
    float*       out     = (float*)d_out;

    // workspace carving (~77 MB)
    float*    xbuf  = (float*)d_ws;                      // [N,128] layer input / layer-0 output
    float*    hbuf  = xbuf + (size_t)NNODES * HID;       // [N,128] h = xW
    float*    asb   = hbuf + (size_t)NNODES * HID;       // [N]
    float*    adb   = asb + NNODES;                      // [N]
    unsigned* mkey  = (unsigned*)(adb + NNODES);         // [N] segment-max keys
    float*    den   = (float*)(mkey + NNODES);           // [N] softmax denominators
    float*    exb   = den + NNODES;                      // [E'] exp(e - m)

    const int GRID_NH = (NNODES * HID + 255) / 256;      // 35000
    const int GRID_N  = (NNODES + 255) / 256;
    const int GRID_E  = (EPLUS + 255) / 256;
    const int GRID_EW = (EPLUS + 7) / 8;                 // wave per edge
    const int GRID_GE = (MTILES + 7) / 8;                // 8 M-tiles per block

    concat_k<<<GRID_NH, 256, 0, stream>>>(users, items, xbuf);

    for (int l = 0; l < 2; ++l) {
        const float* Wl = W + (size_t)l * HID * HID;
        float* tgt = (l == 0) ? xbuf : out;              // layer0 -> scratch, layer1 -> d_out
        gemm_wmma<<<GRID_GE, 256, 0, stream>>>(xbuf, Wl, hbuf);
        alpha_k<<<(NNODES + 7) / 8, 256, 0, stream>>>(hbuf, att_src + l * HID,
                                                      att_dst + l * HID, asb, adb);
        init_nodes<<<GRID_N, 256, 0, stream>>>(mkey, den);
        init_bias<<<GRID_NH, 256, 0, stream>>>(tgt, bias + l * HID);
        edge_max_k<<<GRID_E, 256, 0, stream>>>(ei, asb, adb, mkey);
        edge_exp_k<<<GRID_E, 256, 0, stream>>>(ei, asb, adb, mkey, den, exb);
        edge_agg_k<<<GRID_EW, 256, 0, stream>>>(ei, hbuf, exb, den, tgt);
    }
}